// S2ConvNet_86388972192052
// MI455X (gfx1250) — compile-verified
//
#include <hip/hip_runtime.h>
#include <hip/hip_bf16.h>
#include <math.h>

typedef __attribute__((ext_vector_type(2))) float v2f;
typedef __attribute__((ext_vector_type(8))) float v8f;

// ---------------- problem constants ----------------
#define NTOT 2048
#define CH   256          // n-chunk
#define NCHUNK (NTOT / CH)

__constant__ int   cSQ[6]  = {0, 1, 10, 35, 84, 165};  // sum_{j<l} (2j+1)^2
__constant__ int   cLCO[3] = {0, 1, 10};               // packed (l,mi,pi) offsets, l<3
__constant__ float cC6[6] = {1.f, 0.5f, -0.5f, -1.f, -0.5f, 0.5f};
__constant__ float cS6[6] = {0.f, 0.8660254037844387f, 0.8660254037844387f, 0.f,
                             -0.8660254037844387f, -0.8660254037844387f};

// ---------------- workspace layout (float offsets) ----------------
constexpr size_t O_S2D  = 0;                       // 6 l: 20*(2l+1)   (weighted d_{m,0})
constexpr size_t O_PSI1 = O_S2D  + 720;            // 24*36 complex
constexpr size_t O_Y1C  = O_PSI1 + 864 * 2;        // 8*36  complex  (conj kernel spectrum 1)
constexpr size_t O_D1S  = O_Y1C  + 288 * 2;        // 12 * 286       ((2l+1)*d at b_l1 betas)
constexpr size_t O_EC   = O_D1S  + 3432;           // 12x16 cos   (inverse DFT, e^{+i})
constexpr size_t O_ES   = O_EC   + 192;            // 12x16 sin
constexpr size_t O_ESN  = O_ES   + 192;            // 12x16 -sin
constexpr size_t O_BG   = O_ESN  + 192;            // 12x16 fwd-DFT packed [cos | -sin] cols 0-4|5-9
constexpr size_t O_B2   = O_BG   + 192;            // 12x16 fwd-DFT packed [sin |  cos]
constexpr size_t O_SO3D = O_B2   + 192;            // 12*35          (w*d, l<3)
constexpr size_t O_PSI2 = O_SO3D + 420;            // 144*35 complex
constexpr size_t O_Y2C  = O_PSI2 + 5040 * 2;       // 8*16*35 complex (conj kernel spectrum 2)
constexpr size_t O_D2S  = O_Y2C  + 4480 * 2;       // 6*35           ((2l+1)*d at b_l2 betas)
constexpr size_t O_WINT = O_D2S  + 210;            // 6
constexpr size_t O_FEAT = O_WINT + 6;              // 2048*16 (global, all chunks)
constexpr size_t O_XS   = O_FEAT + (size_t)NTOT * 16;          // CH*36 complex
constexpr size_t NR1    = (size_t)CH * 8 * 12 * 12;            // 294912 GEMM rows
constexpr size_t O_FHRE = O_XS   + (size_t)CH * 72;            // also reused as Ure (fwd FFT)
constexpr size_t O_FHIM = O_FHRE + NR1 * 12;                   // also reused as Uim
constexpr size_t O_TTRE = O_FHIM + NR1 * 12;
constexpr size_t O_TTIM = O_TTRE + NR1 * 12;
constexpr size_t O_H1   = O_TTIM + NR1 * 12;                   // CH*8*12*12*12 real
constexpr size_t O_F2RE = O_H1   + NR1 * 12;
constexpr size_t O_F2IM = O_F2RE + (size_t)CH * 8 * 12 * 25;
constexpr size_t O_XS2  = O_F2IM + (size_t)CH * 8 * 12 * 25;
constexpr size_t O_Z2   = O_XS2  + (size_t)CH * 8 * 35 * 2;
constexpr size_t WS_FLOATS = O_Z2 + (size_t)CH * 16 * 35 * 2;  // ~19.4M floats ~78MB

// ---------------- fp64 table helpers ----------------
__device__ __forceinline__ double lfac(int n) { return lgamma((double)n + 1.0); }

__device__ double wig_d(int l, int mp, int m, double beta) {
  int k0 = max(0, m - mp), k1 = min(l + m, l - mp);
  double c = cos(0.5 * beta), s = sin(0.5 * beta), acc = 0.0;
  for (int k = k0; k <= k1; ++k) {
    double lg = 0.5 * (lfac(l + mp) + lfac(l - mp) + lfac(l + m) + lfac(l - m))
              - lfac(l + m - k) - lfac(k) - lfac(mp - m + k) - lfac(l - mp - k);
    double sgn = ((mp - m + k) & 1) ? -1.0 : 1.0;
    acc += sgn * exp(lg) * pow(c, (double)(2 * l - 2 * k + m - mp))
                         * pow(s, (double)(mp - m + 2 * k));
  }
  return acc;
}
__device__ __forceinline__ double quad_beta(int b, int k) {
  return M_PI * (2 * k + 1) / (4.0 * b);
}
__device__ double quad_w(int b, int k) {
  double beta = quad_beta(b, k), ssum = 0.0;
  for (int j = 0; j < b; ++j) ssum += sin((2 * j + 1) * beta) / (2 * j + 1);
  return (2.0 / b) * sin(beta) * ssum;
}

// ---------------- T1: all data-independent tables ----------------
__global__ void k_tables(float* ws) {
  int gt = blockIdx.x * blockDim.x + threadIdx.x;
  int gs = gridDim.x * blockDim.x;
  // S2 FFT weights: d^l_{m,0}(beta_in_k) * w_in_k
  for (int idx = gt; idx < 6 * 20 * 11; idx += gs) {
    int l = idx / 220, r = idx % 220, k = r / 11, mi = r % 11;
    if (mi >= 2 * l + 1) continue;
    ws[O_S2D + 20 * l * l + k * (2 * l + 1) + mi] =
        (float)(wig_d(l, mi - l, 0, quad_beta(10, k)) * quad_w(10, k));
  }
  // PSI1[g=24][mi] complex
  for (int idx = gt; idx < 6 * 24 * 11; idx += gs) {
    int l = idx / 264, r = idx % 264, g = r / 11, mi = r % 11;
    if (mi >= 2 * l + 1) continue;
    double beta = (g / 8 + 1) * (M_PI / 8.0) / 3.0;
    double alpha = (2.0 * M_PI / 8.0) * (g % 8);
    double d = wig_d(l, mi - l, 0, beta);
    double ang = -(double)(mi - l) * alpha;
    size_t o = O_PSI1 + (size_t)(24 * l * l + g * (2 * l + 1) + mi) * 2;
    ws[o] = (float)(d * cos(ang)); ws[o + 1] = (float)(d * sin(ang));
  }
  // D1s = (2l+1) d^l(beta_l1_k)
  for (int idx = gt; idx < 6 * 12 * 121; idx += gs) {
    int l = idx / (12 * 121), r = idx % (12 * 121);
    int k = r / 121, mi = (r / 11) % 11, ni = r % 11;
    if (mi >= 2 * l + 1 || ni >= 2 * l + 1) continue;
    ws[O_D1S + 12 * cSQ[l] + ((size_t)k * (2 * l + 1) + mi) * (2 * l + 1) + ni] =
        (float)((2 * l + 1) * wig_d(l, mi - l, ni - l, quad_beta(6, k)));
  }
  // 12-pt inverse-DFT matrices e^{+i2pi f a/12} (cols padded to 16)
  for (int idx = gt; idx < 192; idx += gs) {
    int f = idx / 16, a = idx % 16;
    double c = 0, s = 0;
    if (a < 12) { double th = 2.0 * M_PI * f * a / 12.0; c = cos(th); s = sin(th); }
    ws[O_EC + idx] = (float)c; ws[O_ES + idx] = (float)s; ws[O_ESN + idx] = (float)(-s);
  }
  // 12-pt forward-DFT packed matrices: rows f=0..11, cols 0-4 (q=c-2), cols 5-9 (q=c-7)
  // BG = [cos | -sin]  (also serves as B1 of the complex pass), B2 = [sin | cos]
  for (int idx = gt; idx < 192; idx += gs) {
    int f = idx / 16, c = idx % 16;
    double vg = 0, v2 = 0;
    if (c < 10) {
      int q = (c < 5 ? c : c - 5) - 2;
      double th = 2.0 * M_PI * q * f / 12.0;
      if (c < 5) { vg = cos(th);  v2 = sin(th); }
      else       { vg = -sin(th); v2 = cos(th); }
    }
    ws[O_BG + idx] = (float)vg; ws[O_B2 + idx] = (float)v2;
  }
  // SO3 FFT weights: w_l1_k * d^l(beta_l1_k), l<3
  for (int idx = gt; idx < 3 * 12 * 25; idx += gs) {
    int l = idx / 300, r = idx % 300, k = r / 25, mi = (r / 5) % 5, ni = r % 5;
    if (mi >= 2 * l + 1 || ni >= 2 * l + 1) continue;
    ws[O_SO3D + 12 * cSQ[l] + ((size_t)k * (2 * l + 1) + mi) * (2 * l + 1) + ni] =
        (float)(quad_w(6, k) * wig_d(l, mi - l, ni - l, quad_beta(6, k)));
  }
  // PSI2[g=144][mi][ni] complex
  for (int idx = gt; idx < 3 * 144 * 25; idx += gs) {
    int l = idx / 3600, r = idx % 3600, g = r / 25, mi = (r / 5) % 5, ni = r % 5;
    if (mi >= 2 * l + 1 || ni >= 2 * l + 1) continue;
    int bi = g / 48, ai = (g / 6) % 8, gi = g % 6;
    double beta = (bi + 1) * (M_PI / 8.0) / 3.0;
    double alpha = 2.0 * M_PI * ai / 8.0, gamma = 2.0 * M_PI * gi / 6.0;
    double d = wig_d(l, mi - l, ni - l, beta);
    double ang = -(double)(mi - l) * alpha - (double)(ni - l) * gamma;
    size_t o = O_PSI2 + (size_t)(144 * cSQ[l] + (g * (2 * l + 1) + mi) * (2 * l + 1) + ni) * 2;
    ws[o] = (float)(d * cos(ang)); ws[o + 1] = (float)(d * sin(ang));
  }
  // D2s = (2l+1) d^l(beta_l2_k)
  for (int idx = gt; idx < 3 * 6 * 25; idx += gs) {
    int l = idx / 150, r = idx % 150, k = r / 25, mi = (r / 5) % 5, ni = r % 5;
    if (mi >= 2 * l + 1 || ni >= 2 * l + 1) continue;
    ws[O_D2S + 6 * cSQ[l] + ((size_t)k * (2 * l + 1) + mi) * (2 * l + 1) + ni] =
        (float)((2 * l + 1) * wig_d(l, mi - l, ni - l, quad_beta(3, k)));
  }
  for (int idx = gt; idx < 6; idx += gs) ws[O_WINT + idx] = (float)quad_w(3, idx);
}

// ---------------- T2/T3: kernel spectra (depend on w1/w2 inputs) ----------------
__global__ void k_y1(const float* __restrict__ w1, float* ws) {
  int idx = blockIdx.x * blockDim.x + threadIdx.x;
  if (idx >= 6 * 8 * 11) return;
  int l = idx / 88, o = (idx / 11) % 8, pi = idx % 11;
  if (pi >= 2 * l + 1) return;
  float ar = 0.f, ai = 0.f;
  for (int g = 0; g < 24; ++g) {
    float w = w1[o * 24 + g];
    size_t p = O_PSI1 + (size_t)(24 * l * l + g * (2 * l + 1) + pi) * 2;
    ar += w * ws[p]; ai += w * ws[p + 1];
  }
  size_t q = O_Y1C + (size_t)(8 * l * l + o * (2 * l + 1) + pi) * 2;
  ws[q] = ar; ws[q + 1] = -ai;   // conj
}

__global__ void k_y2(const float* __restrict__ w2, float* ws) {
  int idx = blockIdx.x * blockDim.x + threadIdx.x;
  if (idx >= 3 * 8 * 16 * 25) return;
  int l = idx / 3200, r = idx % 3200;
  int i = r / 400, o = (r / 25) % 16, pi = (r / 5) % 5, ki = r % 5;
  if (pi >= 2 * l + 1 || ki >= 2 * l + 1) return;
  float ar = 0.f, ai = 0.f;
  for (int g = 0; g < 144; ++g) {
    float w = w2[(i * 16 + o) * 144 + g];
    size_t p = O_PSI2 + (size_t)(144 * cSQ[l] + (g * (2 * l + 1) + pi) * (2 * l + 1) + ki) * 2;
    ar += w * ws[p]; ai += w * ws[p + 1];
  }
  size_t q = O_Y2C + ((size_t)128 * cSQ[l] + (((size_t)(i * 16 + o) * (2 * l + 1) + pi) * (2 * l + 1) + ki)) * 2;
  ws[q] = ar; ws[q + 1] = -ai;   // conj
}

// ---------------- S2 FFT: x -> xs[l][n,m] ----------------
__global__ void k_s2x(const float* __restrict__ x, float* ws, int n0) {
  int idx = blockIdx.x * blockDim.x + threadIdx.x;
  if (idx >= CH * 11) return;
  int nl = idx / 11, m = idx % 11 - 5, am = abs(m);
  const float* xp = x + (size_t)(n0 + nl) * 400;
  float accr[6] = {0}, acci[6] = {0};
  const float w0 = -6.283185307179586f / 20.0f;
  for (int k = 0; k < 20; ++k) {
    float fr = 0.f, fi = 0.f;
    for (int a = 0; a < 20; ++a) {
      float s, c; __sincosf(w0 * (float)(a * m), &s, &c);
      float v = xp[k * 20 + a];
      fr += v * c; fi += v * s;
    }
#pragma unroll
    for (int l = 0; l < 6; ++l)
      if (l >= am) {
        float d = ws[O_S2D + 20 * l * l + k * (2 * l + 1) + (m + l)];
        accr[l] += fr * d; acci[l] += fi * d;
      }
  }
#pragma unroll
  for (int l = 0; l < 6; ++l)
    if (l >= am) {
      size_t o = O_XS + (size_t)nl * 72 + (size_t)(l * l + m + l) * 2;
      ws[o] = accr[l]; ws[o + 1] = acci[l];
    }
}

// ---------------- Fh[(n,o,k,pu), mu] = sum_l xs * conj(y1) * (2l+1) d ----------------
__global__ void k_fh(float* ws) {
  size_t idx = (size_t)blockIdx.x * blockDim.x + threadIdx.x;
  if (idx >= (size_t)CH * 8 * 12 * 144) return;
  int mu = idx % 12; size_t t = idx / 12;
  int pu = t % 12; t /= 12;
  int k = t % 12; t /= 12;
  int o = t % 8; int nl = (int)(t / 8);
  int m = (mu <= 5) ? mu : mu - 12;
  int p = (pu <= 5) ? pu : pu - 12;
  float fr = 0.f, fi = 0.f;
  if (m >= -5 && m <= 5 && p >= -5 && p <= 5) {   // mu/pu == 6 -> zero row/col
    int lmin = max(abs(m), abs(p));
    for (int l = lmin; l < 6; ++l) {
      size_t xo = O_XS + (size_t)nl * 72 + (size_t)(l * l + m + l) * 2;
      float xr = ws[xo], xi = ws[xo + 1];
      size_t yo = O_Y1C + (size_t)(8 * l * l + o * (2 * l + 1) + (p + l)) * 2;
      float yr = ws[yo], yi = ws[yo + 1];
      float d = ws[O_D1S + 12 * cSQ[l] + ((size_t)k * (2 * l + 1) + (m + l)) * (2 * l + 1) + (p + l)];
      fr += d * (xr * yr - xi * yi);
      fi += d * (xr * yi + xi * yr);
    }
  }
  size_t row = ((size_t)((nl * 8 + o) * 12 + k) * 12 + pu);
  ws[O_FHRE + row * 12 + mu] = fr;
  ws[O_FHIM + row * 12 + mu] = fi;
}

// ---------------- GEMM1 (WMMA f32): contract mu -> alpha, transpose-store ----------------
__global__ __launch_bounds__(256) void k_gemm1(float* ws) {
  const float* fre = ws + O_FHRE;
  const float* fim = ws + O_FHIM;
  float* tre = ws + O_TTRE;
  float* tim = ws + O_TTIM;
  int wave = threadIdx.x >> 5, lane = threadIdx.x & 31;
  int tile = blockIdx.x * 8 + wave;
  int rowBase = tile * 16;
  int col = lane & 15, hi = lane >> 4;
  v2f Bc[3], Bs[3], Bn[3];
#pragma unroll
  for (int kb = 0; kb < 3; ++kb) {
    int K0 = 4 * kb + 2 * hi;
    Bc[kb] = (v2f){ws[O_EC + K0 * 16 + col],  ws[O_EC + (K0 + 1) * 16 + col]};
    Bs[kb] = (v2f){ws[O_ES + K0 * 16 + col],  ws[O_ES + (K0 + 1) * 16 + col]};
    Bn[kb] = (v2f){ws[O_ESN + K0 * 16 + col], ws[O_ESN + (K0 + 1) * 16 + col]};
  }
  const float* rr = fre + (size_t)(rowBase + col) * 12;
  const float* ri = fim + (size_t)(rowBase + col) * 12;
  __builtin_prefetch(rr + 8 * 16 * 12, 0, 0);
  __builtin_prefetch(ri + 8 * 16 * 12, 0, 0);
  v8f accR = {}; v8f accI = {};
#pragma unroll
  for (int kb = 0; kb < 3; ++kb) {
    int K0 = 4 * kb + 2 * hi;
    v2f aR = (v2f){rr[K0], rr[K0 + 1]};
    v2f aI = (v2f){ri[K0], ri[K0 + 1]};
    accR = __builtin_amdgcn_wmma_f32_16x16x4_f32(false, aR, false, Bc[kb], (short)0, accR, false, false);
    accR = __builtin_amdgcn_wmma_f32_16x16x4_f32(false, aI, false, Bn[kb], (short)0, accR, false, false);
    accI = __builtin_amdgcn_wmma_f32_16x16x4_f32(false, aR, false, Bs[kb], (short)0, accI, false, false);
    accI = __builtin_amdgcn_wmma_f32_16x16x4_f32(false, aI, false, Bc[kb], (short)0, accI, false, false);
  }
  if (col < 12) {
#pragma unroll
    for (int j = 0; j < 8; ++j) {
      int r = rowBase + j + 8 * hi;           // (nok, pu) flattened
      int nok = r / 12, pu = r - nok * 12;
      size_t oidx = ((size_t)nok * 12 + col) * 12 + pu;   // -> (nok, alpha, pu)
      tre[oidx] = accR[j]; tim[oidx] = accI[j];
    }
  }
}

// ---------------- GEMM2 (WMMA f32): contract pu -> gamma, real part + bias + ReLU ----------
__global__ __launch_bounds__(256) void k_gemm2(float* ws, const float* __restrict__ b1) {
  const float* rrb = ws + O_TTRE;
  const float* rib = ws + O_TTIM;
  float* h1 = ws + O_H1;
  int wave = threadIdx.x >> 5, lane = threadIdx.x & 31;
  int tile = blockIdx.x * 8 + wave;
  int rowBase = tile * 16;
  int col = lane & 15, hi = lane >> 4;
  v2f Bc[3], Bn[3];
#pragma unroll
  for (int kb = 0; kb < 3; ++kb) {
    int K0 = 4 * kb + 2 * hi;
    Bc[kb] = (v2f){ws[O_EC + K0 * 16 + col],  ws[O_EC + (K0 + 1) * 16 + col]};
    Bn[kb] = (v2f){ws[O_ESN + K0 * 16 + col], ws[O_ESN + (K0 + 1) * 16 + col]};
  }
  const float* rr = rrb + (size_t)(rowBase + col) * 12;
  const float* ri = rib + (size_t)(rowBase + col) * 12;
  __builtin_prefetch(rr + 8 * 16 * 12, 0, 0);
  __builtin_prefetch(ri + 8 * 16 * 12, 0, 0);
  v8f acc = {};
#pragma unroll
  for (int kb = 0; kb < 3; ++kb) {
    int K0 = 4 * kb + 2 * hi;
    v2f aR = (v2f){rr[K0], rr[K0 + 1]};
    v2f aI = (v2f){ri[K0], ri[K0 + 1]};
    acc = __builtin_amdgcn_wmma_f32_16x16x4_f32(false, aR, false, Bc[kb], (short)0, acc, false, false);
    acc = __builtin_amdgcn_wmma_f32_16x16x4_f32(false, aI, false, Bn[kb], (short)0, acc, false, false);
  }
  if (col < 12) {
#pragma unroll
    for (int j = 0; j < 8; ++j) {
      int r = rowBase + j + 8 * hi;           // (nok, alpha) flattened
      int nok = r / 12;
      int o = (nok / 12) & 7;
      float v = acc[j] + b1[o];
      h1[(size_t)r * 12 + col] = fmaxf(v, 0.f);
    }
  }
}

// ------- SO3 forward FFT, gamma pass (WMMA f32): real h1 -> packed [Re|Im], transpose -------
__global__ __launch_bounds__(256) void k_so3fa(float* ws) {
  const float* h1 = ws + O_H1;
  float* ure = ws + O_FHRE;   // reuse dead Fh buffers: [(nok*5+p5)*12 + A]
  float* uim = ws + O_FHIM;
  int wave = threadIdx.x >> 5, lane = threadIdx.x & 31;
  int tile = blockIdx.x * 8 + wave;
  int rowBase = tile * 16;                     // rows: (nok, A)
  int col = lane & 15, hi = lane >> 4;
  v2f Bg[3];
#pragma unroll
  for (int kb = 0; kb < 3; ++kb) {
    int K0 = 4 * kb + 2 * hi;
    Bg[kb] = (v2f){ws[O_BG + K0 * 16 + col], ws[O_BG + (K0 + 1) * 16 + col]};
  }
  const float* rp = h1 + (size_t)(rowBase + col) * 12;
  __builtin_prefetch(rp + 8 * 16 * 12, 0, 0);
  v8f acc = {};
#pragma unroll
  for (int kb = 0; kb < 3; ++kb) {
    int K0 = 4 * kb + 2 * hi;
    v2f aH = (v2f){rp[K0], rp[K0 + 1]};
    acc = __builtin_amdgcn_wmma_f32_16x16x4_f32(false, aH, false, Bg[kb], (short)0, acc, false, false);
  }
  if (col < 10) {
    int p5 = (col < 5) ? col : col - 5;
    float* dst = (col < 5) ? ure : uim;
#pragma unroll
    for (int j = 0; j < 8; ++j) {
      int r = rowBase + j + 8 * hi;
      int nok = r / 12, A = r - nok * 12;
      dst[((size_t)nok * 5 + p5) * 12 + A] = acc[j];
    }
  }
}

// ------- SO3 forward FFT, alpha pass (WMMA f32): complex U -> F2 coefficients -------
__global__ __launch_bounds__(256) void k_so3fb(float* ws) {
  const float* ure = ws + O_FHRE;
  const float* uim = ws + O_FHIM;
  float* f2re = ws + O_F2RE;
  float* f2im = ws + O_F2IM;
  int wave = threadIdx.x >> 5, lane = threadIdx.x & 31;
  int tile = blockIdx.x * 8 + wave;
  int rowBase = tile * 16;                     // rows: (nok, p5)
  int col = lane & 15, hi = lane >> 4;
  v2f Bg[3], B2[3];
#pragma unroll
  for (int kb = 0; kb < 3; ++kb) {
    int K0 = 4 * kb + 2 * hi;
    Bg[kb] = (v2f){ws[O_BG + K0 * 16 + col], ws[O_BG + (K0 + 1) * 16 + col]};
    B2[kb] = (v2f){ws[O_B2 + K0 * 16 + col], ws[O_B2 + (K0 + 1) * 16 + col]};
  }
  const float* rr = ure + (size_t)(rowBase + col) * 12;
  const float* ri = uim + (size_t)(rowBase + col) * 12;
  __builtin_prefetch(rr + 8 * 16 * 12, 0, 0);
  __builtin_prefetch(ri + 8 * 16 * 12, 0, 0);
  v8f acc = {};
#pragma unroll
  for (int kb = 0; kb < 3; ++kb) {
    int K0 = 4 * kb + 2 * hi;
    v2f aR = (v2f){rr[K0], rr[K0 + 1]};
    v2f aI = (v2f){ri[K0], ri[K0 + 1]};
    // Re = Ure*cos + Uim*sin ; Im = Uim*cos - Ure*sin  (packed in cols 0-4 | 5-9)
    acc = __builtin_amdgcn_wmma_f32_16x16x4_f32(false, aR, false, Bg[kb], (short)0, acc, false, false);
    acc = __builtin_amdgcn_wmma_f32_16x16x4_f32(false, aI, false, B2[kb], (short)0, acc, false, false);
  }
  if (col < 10) {
    int m5 = (col < 5) ? col : col - 5;
    float* dst = (col < 5) ? f2re : f2im;
#pragma unroll
    for (int j = 0; j < 8; ++j) {
      int r = rowBase + j + 8 * hi;
      int nok = r / 5, p5 = r - nok * 5;
      dst[(size_t)nok * 25 + m5 * 5 + p5] = acc[j];
    }
  }
}

// ---------------- xs2[l][n,i,m,p] = sum_k F2 * (w*d) ----------------
__global__ void k_xs2(float* ws) {
  int idx = blockIdx.x * blockDim.x + threadIdx.x;
  if (idx >= CH * 8 * 35) return;
  int lc = idx % 35, i = (idx / 35) % 8, nl = idx / 280;
  int l, mi, pi;
  if (lc < 1)       { l = 0; mi = 0; pi = 0; }
  else if (lc < 10) { l = 1; mi = (lc - 1) / 3; pi = (lc - 1) % 3; }
  else              { l = 2; mi = (lc - 10) / 5; pi = (lc - 10) % 5; }
  int fidx = (mi - l + 2) * 5 + (pi - l + 2);
  float ar = 0.f, ai = 0.f;
  for (int k = 0; k < 12; ++k) {
    float d = ws[O_SO3D + 12 * cSQ[l] + ((size_t)k * (2 * l + 1) + mi) * (2 * l + 1) + pi];
    size_t fo = (size_t)((nl * 8 + i) * 12 + k) * 25 + fidx;
    ar += ws[O_F2RE + fo] * d; ai += ws[O_F2IM + fo] * d;
  }
  size_t o = O_XS2 + ((size_t)(nl * 8 + i) * 35 + lc) * 2;
  ws[o] = ar; ws[o + 1] = ai;
}

// ---------------- z2[n,o,m,p] = sum_{i,k} xs2 * conj(y2) ----------------
__global__ void k_z2(float* ws) {
  int idx = blockIdx.x * blockDim.x + threadIdx.x;
  if (idx >= CH * 16 * 35) return;
  int lc = idx % 35, o = (idx / 35) % 16, nl = idx / (16 * 35);
  int l, mi, pi;
  if (lc < 1)       { l = 0; mi = 0; pi = 0; }
  else if (lc < 10) { l = 1; mi = (lc - 1) / 3; pi = (lc - 1) % 3; }
  else              { l = 2; mi = (lc - 10) / 5; pi = (lc - 10) % 5; }
  float ar = 0.f, ai = 0.f;
  for (int i = 0; i < 8; ++i)
    for (int ki = 0; ki < 2 * l + 1; ++ki) {
      int lc2 = cLCO[l] + mi * (2 * l + 1) + ki;
      size_t xo = O_XS2 + ((size_t)(nl * 8 + i) * 35 + lc2) * 2;
      float xr = ws[xo], xi = ws[xo + 1];
      size_t yo = O_Y2C + ((size_t)128 * cSQ[l] +
                  (((size_t)(i * 16 + o) * (2 * l + 1) + pi) * (2 * l + 1) + ki)) * 2;
      float yr = ws[yo], yi = ws[yo + 1];
      ar += xr * yr - xi * yi; ai += xr * yi + xi * yr;
    }
  size_t zo = O_Z2 + ((size_t)(nl * 16 + o) * 35 + lc) * 2;
  ws[zo] = ar; ws[zo + 1] = ai;
}

// ---------------- b=3 inverse FFT + bias + ReLU + SO3 integrate ----------------
__global__ void k_h2int(float* ws, const float* __restrict__ b2, int n0) {
  int idx = blockIdx.x * blockDim.x + threadIdx.x;
  if (idx >= CH * 16) return;
  int nl = idx / 16, o = idx % 16;
  float bb = b2[o];
  const float* zz = ws + O_Z2 + (size_t)(nl * 16 + o) * 70;
  float acc = 0.f;
  for (int k = 0; k < 6; ++k) {
    float tre[25], tim[25];
#pragma unroll
    for (int m5 = 0; m5 < 5; ++m5)
#pragma unroll
      for (int p5 = 0; p5 < 5; ++p5) {
        int m = m5 - 2, p = p5 - 2;
        float sr = 0.f, si = 0.f;
        int lmin = max(abs(m), abs(p));
#pragma unroll
        for (int l = 0; l < 3; ++l)
          if (l >= lmin) {
            int lc = cLCO[l] + (m + l) * (2 * l + 1) + (p + l);
            float d = ws[O_D2S + 6 * cSQ[l] + ((size_t)k * (2 * l + 1) + (m + l)) * (2 * l + 1) + (p + l)];
            sr += zz[lc * 2] * d; si += zz[lc * 2 + 1] * d;
          }
        tre[m5 * 5 + p5] = sr; tim[m5 * 5 + p5] = si;
      }
    float s1 = 0.f;
    for (int A = 0; A < 6; ++A)
      for (int G = 0; G < 6; ++G) {
        float h = 0.f;
#pragma unroll
        for (int m5 = 0; m5 < 5; ++m5)
#pragma unroll
          for (int p5 = 0; p5 < 5; ++p5) {
            int jj = ((m5 - 2) * A + (p5 - 2) * G + 24) % 6;
            h += tre[m5 * 5 + p5] * cC6[jj] - tim[m5 * 5 + p5] * cS6[jj];
          }
        s1 += fmaxf(h + bb, 0.f);
      }
    acc += ws[O_WINT + k] * s1 * (1.0f / 36.0f);
  }
  ws[O_FEAT + (size_t)(n0 + nl) * 16 + o] = acc;
}

// ---------------- max-pool over 512 points + linear head ----------------
__global__ void k_final(const float* __restrict__ w_out, const float* __restrict__ bias_out,
                        const float* __restrict__ ws, float* __restrict__ out) {
  __shared__ float mx[64];
  int tid = threadIdx.x;
  if (tid < 64) {
    int b = tid >> 4, o = tid & 15;
    const float* fp = ws + O_FEAT + (size_t)b * 512 * 16 + o;
    float m = -INFINITY;
    for (int p = 0; p < 512; ++p) m = fmaxf(m, fp[(size_t)p * 16]);
    mx[tid] = m;
  }
  __syncthreads();
  if (tid < 40) {
    int b = tid / 10, j = tid % 10;
    float s = bias_out[j];
    for (int o = 0; o < 16; ++o) s += w_out[j * 16 + o] * mx[b * 16 + o];
    out[b * 10 + j] = s;
  }
}

extern "C" void kernel_launch(void* const* d_in, const int* in_sizes, int n_in,
                              void* d_out, int out_size, void* d_ws, size_t ws_size,
                              hipStream_t stream) {
  (void)in_sizes; (void)n_in; (void)out_size; (void)ws_size;
  const float* x        = (const float*)d_in[0];
  const float* w1       = (const float*)d_in[1];
  const float* b1       = (const float*)d_in[2];
  const float* w2       = (const float*)d_in[3];
  const float* b2       = (const float*)d_in[4];
  const float* w_out    = (const float*)d_in[5];
  const float* bias_out = (const float*)d_in[6];
  float* out = (float*)d_out;
  float* ws  = (float*)d_ws;

  k_tables<<<96, 128, 0, stream>>>(ws);
  k_y1<<<3, 256, 0, stream>>>(w1, ws);
  k_y2<<<38, 256, 0, stream>>>(w2, ws);

  for (int c = 0; c < NCHUNK; ++c) {
    int n0 = c * CH;
    k_s2x<<<(CH * 11 + 255) / 256, 256, 0, stream>>>(x, ws, n0);
    k_fh<<<(int)(((size_t)CH * 8 * 12 * 144) / 256), 256, 0, stream>>>(ws);
    k_gemm1<<<(int)(NR1 / 16 / 8), 256, 0, stream>>>(ws);
    k_gemm2<<<(int)(NR1 / 16 / 8), 256, 0, stream>>>(ws, b1);
    // forward SO3 FFT as two row-batched WMMA DFT passes
    k_so3fa<<<(int)(NR1 / 16 / 8), 256, 0, stream>>>(ws);               // rows (nok, A)
    k_so3fb<<<(int)(((size_t)CH * 8 * 12 * 5) / 16 / 8), 256, 0, stream>>>(ws); // rows (nok, p5)
    k_xs2<<<(CH * 8 * 35 + 255) / 256, 256, 0, stream>>>(ws);
    k_z2<<<(CH * 16 * 35 + 255) / 256, 256, 0, stream>>>(ws);
    k_h2int<<<(CH * 16 + 255) / 256, 256, 0, stream>>>(ws, b2, n0);
  }
  k_final<<<1, 128, 0, stream>>>(w_out, bias_out, ws, out);
}